// ODEFunc_81570018885911
// MI455X (gfx1250) — compile-verified
//
#include <hip/hip_runtime.h>
#include <hip/hip_bf16.h>
#include <math.h>

typedef float v2f __attribute__((ext_vector_type(2)));
typedef float v8f __attribute__((ext_vector_type(8)));

#define ZSTR 66          // z-tile row stride in floats (even => b64 LDS reads; odd/64 => no bank conflicts)
#define TILES_PER_WAVE 4 // grid-stride amortization of constant-operand setup

__device__ __forceinline__ float fast_tanh(float x) {
#if __has_builtin(__builtin_amdgcn_tanhf)
    return __builtin_amdgcn_tanhf(x);          // gfx1250 v_tanh_f32
#elif __has_builtin(__builtin_amdgcn_tanh_f32)
    return __builtin_amdgcn_tanh_f32(x);
#else
    return tanhf(x);
#endif
}

// ---------------------------------------------------------------------------
// Kernel 1: tiny QP fixup from row 0 -> W2n (3x50) into workspace
// ---------------------------------------------------------------------------
__global__ void fixup_kernel(const float* __restrict__ y,
                             const float* __restrict__ W1,
                             const float* __restrict__ b1,
                             const float* __restrict__ W2,
                             const float* __restrict__ b2,
                             float* __restrict__ w2n) {
    __shared__ float z0[50];
    int tid = threadIdx.x;
    if (tid < 50) {
        float s = b1[tid];
        #pragma unroll
        for (int k = 0; k < 3; ++k) s += W1[tid * 3 + k] * y[k];
        z0[tid] = tanhf(s);
    }
    __syncthreads();
    if (tid == 0) {
        float bias0[3];
        for (int i = 0; i < 3; ++i) {
            float s = b2[i];
            for (int j = 3; j < 50; ++j) s += W2[i * 50 + j] * z0[j];
            bias0[i] = s;
        }
        float b_cbf = 0.5f * (y[0] + y[1]) - y[2];
        float Lfb   = 0.5f * (bias0[0] + bias0[1]) - bias0[2];
        float h     = Lfb + b_cbf;

        float g[9], xu[9];
        for (int i = 0; i < 3; ++i) {
            float zi = z0[i];
            g[3 * i + 0] = -0.5f * zi;
            g[3 * i + 1] = -0.5f * zi;
            g[3 * i + 2] =  zi;
        }
        // x_unc = W2[:, :3].T flattened:  xu[3i+j] = W2[j][i]
        for (int i = 0; i < 3; ++i)
            for (int j = 0; j < 3; ++j)
                xu[3 * i + j] = W2[j * 50 + i];

        float gx = 0.f, gg = 0.f;
        for (int i = 0; i < 9; ++i) { gx += g[i] * xu[i]; gg += g[i] * g[i]; }
        float lam = fmaxf((gx - h) / (gg + 1e-12f), 0.0f);

        float x[9];
        for (int i = 0; i < 9; ++i) x[i] = xu[i] - lam * g[i];

        // W2n[i][j] = x[3j+i] for j<3, else W2[i][j]
        for (int i = 0; i < 3; ++i)
            for (int j = 0; j < 50; ++j)
                w2n[i * 50 + j] = (j < 3) ? x[3 * j + i] : W2[i * 50 + j];
    }
}

// ---------------------------------------------------------------------------
// Kernel 2: bulk pass. One wave32 per 16-row tile, grid-striding over tiles.
//   z = tanh(y @ W1^T + b1)   via V_WMMA_F32_16X16X4_F32 (4 N-tiles, K=3->4)
//   out = z @ W2n^T + b2      via V_WMMA_F32_16X16X4_F32 (13 K-steps over 50->52)
// All constant WMMA operands (W1^T tiles, W2n^T chunks, biases: 40 VGPRs) are
// built ONCE per wave; per-tile work is just loads + 17 WMMAs + tanh + LDS
// transpose. z goes through LDS (D-layout -> A-layout).
// ---------------------------------------------------------------------------
__global__ __launch_bounds__(256) void ode_bulk_kernel(
        const float* __restrict__ y,
        const float* __restrict__ W1,
        const float* __restrict__ b1,
        const float* __restrict__ b2,
        const float* __restrict__ w2n,
        float* __restrict__ out,
        int numTiles) {
    __shared__ float w2s[152];
    __shared__ float zls[8][16 * ZSTR];   // per-wave z tile, 16 rows x 66 floats

    int tid = threadIdx.x;
    if (tid < 150) w2s[tid] = w2n[tid];
    __syncthreads();

    int wave = tid >> 5;
    int lane = tid & 31;
    int half = lane >> 4;                  // 0: lanes 0-15, 1: lanes 16-31
    int ln   = lane & 15;

    // ---- loop-invariant B operand for GEMM1: W1^T, 4x16 tiles (K padded 3->4) ----
    // f32 4x16 B-layout: V0 = (K=0 | K=2), V1 = (K=1 | K=3); N = lane%16
    v2f bW[4];
    float b1v[4];
    #pragma unroll
    for (int tt = 0; tt < 4; ++tt) {
        int n = tt * 16 + ln;
        float v0 = 0.f, v1 = 0.f, bb = 0.f;
        if (n < 50) {
            v0 = half ? W1[n * 3 + 2] : W1[n * 3 + 0];
            v1 = half ? 0.f           : W1[n * 3 + 1];   // K=3 is zero pad
            bb = b1[n];
        }
        bW[tt].x = v0; bW[tt].y = v1;
        b1v[tt] = bb;
    }

    // ---- loop-invariant B operand for GEMM2: W2n^T chunks, 13 K=4 steps ----
    v2f bB[13];
    #pragma unroll
    for (int s = 0; s < 13; ++s) {
        int k = 4 * s + 2 * half;
        float v0 = 0.f, v1 = 0.f;
        if (ln < 3) {
            if (k     < 50) v0 = w2s[ln * 50 + k];
            if (k + 1 < 50) v1 = w2s[ln * 50 + k + 1];
        }
        bB[s].x = v0; bB[s].y = v1;
    }
    float b2v = (ln < 3) ? b2[ln] : 0.f;

    float* zl = &zls[wave][0];
    const v2f* z2 = (const v2f*)zl;

    int waveStride = gridDim.x * 8;        // total wave32s in grid
    for (int tile = blockIdx.x * 8 + wave; tile < numTiles; tile += waveStride) {
        int rowbase = tile * 16;
        int m = rowbase + ln;

        // prefetch next grid-stride iteration's y rows (global_prefetch_b8)
        if (tile + waveStride < numTiles) {
            __builtin_prefetch(&y[3 * (m + waveStride * 16)], 0, 1);
        }

        // ---- phase A: z = tanh(y @ W1^T + b1), 16 rows x 64 (cols 50..63 -> 0) ----
        // f32 16x4 A-layout: V0 = (K=0 | K=2), V1 = (K=1 | K=3); M = lane%16
        v2f aY;
        aY.x = y[3 * m + 2 * half];
        float y1 = y[3 * m + 1];
        aY.y = half ? 0.f : y1;

        #pragma unroll
        for (int tt = 0; tt < 4; ++tt) {
            v8f acc = {};
            acc = __builtin_amdgcn_wmma_f32_16x16x4_f32(
                      false, aY, false, bW[tt], (short)0, acc, false, false);
            // D-layout: VGPR j holds (row = j + 8*half, col = tt*16 + lane%16)
            int col = tt * 16 + ln;
            #pragma unroll
            for (int j = 0; j < 8; ++j) {
                float zv = fast_tanh(acc[j] + b1v[tt]);
                int row = j + 8 * half;
                zl[row * ZSTR + col] = zv;   // fuses into ds_store_2addr_b32
            }
        }

        // ---- phase B: out = z @ W2n^T + b2, 13 WMMA K-steps ----
        v8f acc2 = {};
        #pragma unroll
        for (int s = 0; s < 13; ++s) {
            // A-layout read: lane -> (M = lane%16, K = 4s + 2*half .. +1) => ds_load_b64
            v2f a = z2[ln * (ZSTR / 2) + 2 * s + half];
            acc2 = __builtin_amdgcn_wmma_f32_16x16x4_f32(
                       false, a, false, bB[s], (short)0, acc2, false, false);
        }

        // D-layout store: VGPR j -> out[rowbase + j + 8*half][N=lane%16], N<3 valid
        #pragma unroll
        for (int j = 0; j < 8; ++j) {
            if (ln < 3) {
                int row = rowbase + j + 8 * half;
                out[row * 3 + ln] = acc2[j] + b2v;
            }
        }
    }
}

// ---------------------------------------------------------------------------
// Defensive scalar tail for N % 16 != 0 (dead for N = 1e6)
// ---------------------------------------------------------------------------
__global__ void ode_tail_kernel(const float* __restrict__ y,
                                const float* __restrict__ W1,
                                const float* __restrict__ b1,
                                const float* __restrict__ b2,
                                const float* __restrict__ w2n,
                                float* __restrict__ out,
                                int start, int N) {
    int r = start + blockIdx.x * blockDim.x + threadIdx.x;
    if (r >= N) return;
    float y0 = y[3 * r], y1 = y[3 * r + 1], y2 = y[3 * r + 2];
    float o[3] = {b2[0], b2[1], b2[2]};
    for (int j = 0; j < 50; ++j) {
        float z = tanhf(W1[j * 3] * y0 + W1[j * 3 + 1] * y1 + W1[j * 3 + 2] * y2 + b1[j]);
        o[0] += w2n[0 * 50 + j] * z;
        o[1] += w2n[1 * 50 + j] * z;
        o[2] += w2n[2 * 50 + j] * z;
    }
    out[3 * r + 0] = o[0];
    out[3 * r + 1] = o[1];
    out[3 * r + 2] = o[2];
}

extern "C" void kernel_launch(void* const* d_in, const int* in_sizes, int n_in,
                              void* d_out, int out_size, void* d_ws, size_t ws_size,
                              hipStream_t stream) {
    // inputs: t(1), y(N*3), W1(150), b1(50), W2(150), b2(3)
    const float* y  = (const float*)d_in[1];
    const float* W1 = (const float*)d_in[2];
    const float* b1 = (const float*)d_in[3];
    const float* W2 = (const float*)d_in[4];
    const float* b2 = (const float*)d_in[5];
    float* out = (float*)d_out;
    float* w2n = (float*)d_ws;   // 150 floats of scratch

    int N = in_sizes[1] / 3;
    int numTiles = N / 16;
    int tailStart = numTiles * 16;
    int tail = N - tailStart;

    fixup_kernel<<<1, 64, 0, stream>>>(y, W1, b1, W2, b2, w2n);

    if (numTiles > 0) {
        // 8 wave32 tiles per 256-thread block; each wave grid-strides over
        // ~TILES_PER_WAVE tiles to amortize the 40-VGPR constant-operand setup.
        int blocks = (numTiles + 8 * TILES_PER_WAVE - 1) / (8 * TILES_PER_WAVE);
        ode_bulk_kernel<<<blocks, 256, 0, stream>>>(y, W1, b1, b2, w2n, out, numTiles);
    }
    if (tail > 0) {
        int blocks = (tail + 63) / 64;
        ode_tail_kernel<<<blocks, 64, 0, stream>>>(y, W1, b1, b2, w2n, out, tailStart, N);
    }
}